// SparseGraphAttentionLayer_69441031242184
// MI455X (gfx1250) — compile-verified
//
#include <hip/hip_runtime.h>
#include <hip/hip_bf16.h>

typedef float v2f __attribute__((ext_vector_type(2)));
typedef float v8f __attribute__((ext_vector_type(8)));

#define IN_F 128
#define OUT_F 64
#define NEG_SLOPE 0.2f
#define WT_STRIDE 132   // padded transposed-W stride: distinct LDS banks for all 32 lanes

// ---------- helpers: order-preserving float<->uint map for atomic max ----------
__device__ __forceinline__ unsigned f2ord(float f) {
    unsigned u = __float_as_uint(f);
    return (u & 0x80000000u) ? ~u : (u | 0x80000000u);
}
__device__ __forceinline__ float ord2f(unsigned u) {
    return (u & 0x80000000u) ? __uint_as_float(u & 0x7FFFFFFFu)
                             : __uint_as_float(~u);
}

// ---------------------------------------------------------------------------
// Kernel 1: Wh = x @ W using V_WMMA_F32_16X16X4_F32.
// One wave = one 16-row tile, 4 column tiles (64 cols), K=128 in steps of 4.
// A layout (32-bit 16x4): lanes 0-15 -> M, half-wave selects K{0,1} vs K{2,3}.
// B layout (32-bit 4x16): VGPR0 = rows K{0|2}, VGPR1 = rows K{1|3}, N = lane%16.
// W is staged TRANSPOSED in LDS (lWt[col][k], stride 132) so each B fragment
// is one 8-byte contiguous pair -> single ds_load_b64, no register shuffles.
// ---------------------------------------------------------------------------
__global__ __launch_bounds__(256) void gat_gemm_wmma(
    const float* __restrict__ x, const float* __restrict__ W,
    float* __restrict__ Wh, int n)
{
    __shared__ float lWt[OUT_F * WT_STRIDE]; // 33 KB, transposed + padded
    int tid = threadIdx.x;

    // stage W transposed: lWt[c*132 + k] = W[k*64 + c] (coalesced global reads)
#pragma unroll
    for (int i = 0; i < (IN_F * OUT_F) / 256; ++i) {
        int idx = i * 256 + tid;          // linear over W
        int k = idx >> 6;                 // /64
        int c = idx & 63;
        lWt[c * WT_STRIDE + k] = W[idx];
    }
    __syncthreads();

    int wave = tid >> 5;
    int lane = tid & 31;
    int half = lane >> 4;   // 0: K{0,1} / rows 0-7 of D, 1: K{2,3} / rows 8-15
    int r    = lane & 15;   // M index for A, N index for B/D
    int row_base = blockIdx.x * 128 + wave * 16;

    int grow = row_base + r;
    if (grow > n - 1) grow = n - 1;               // clamp loads; stores guarded
    const float* xrow = x + (size_t)grow * IN_F + 2 * half;

    // per-lane base into transposed W: column tile t adds 16*WT_STRIDE
    const float* wcol = lWt + r * WT_STRIDE + 2 * half;

    v8f acc0 = {}, acc1 = {}, acc2 = {}, acc3 = {};

#pragma unroll
    for (int kc = 0; kc < IN_F / 4; ++kc) {
        int k = kc * 4;
        v2f a  = *(const v2f*)(xrow + k);          // x[row][k+2h], x[row][k+2h+1]
        v2f b0 = *(const v2f*)(wcol + k + 0 * 16 * WT_STRIDE);
        v2f b1 = *(const v2f*)(wcol + k + 1 * 16 * WT_STRIDE);
        v2f b2 = *(const v2f*)(wcol + k + 2 * 16 * WT_STRIDE);
        v2f b3 = *(const v2f*)(wcol + k + 3 * 16 * WT_STRIDE);
        acc0 = __builtin_amdgcn_wmma_f32_16x16x4_f32(false, a, false, b0, (short)0, acc0, false, false);
        acc1 = __builtin_amdgcn_wmma_f32_16x16x4_f32(false, a, false, b1, (short)0, acc1, false, false);
        acc2 = __builtin_amdgcn_wmma_f32_16x16x4_f32(false, a, false, b2, (short)0, acc2, false, false);
        acc3 = __builtin_amdgcn_wmma_f32_16x16x4_f32(false, a, false, b3, (short)0, acc3, false, false);
    }

    // D layout: VGPR v -> row (half*8 + v), N = lane%16 within each 16-col tile
    int row0 = row_base + half * 8;
#pragma unroll
    for (int v = 0; v < 8; ++v) {
        int row = row0 + v;
        if (row < n) {
            float* o = Wh + (size_t)row * OUT_F + r;
            o[0]  = acc0[v];
            o[16] = acc1[v];
            o[32] = acc2[v];
            o[48] = acc3[v];
        }
    }
}

// ---------------------------------------------------------------------------
// Kernel 2: per-node attention scores s_src = Wh.a_src, s_dst = Wh.a_dst
// one wave per row, wave32 shuffle reduction
// ---------------------------------------------------------------------------
__global__ __launch_bounds__(256) void gat_scores(
    const float* __restrict__ Wh, const float* __restrict__ a_src,
    const float* __restrict__ a_dst, float* __restrict__ s_src,
    float* __restrict__ s_dst, int n)
{
    int gid = blockIdx.x * blockDim.x + threadIdx.x;
    int row = gid >> 5;
    int lane = gid & 31;
    if (row >= n) return;

    v2f w  = *(const v2f*)(Wh + (size_t)row * OUT_F + 2 * lane);
    v2f as = *(const v2f*)(a_src + 2 * lane);
    v2f ad = *(const v2f*)(a_dst + 2 * lane);
    float p = w.x * as.x + w.y * as.y;
    float q = w.x * ad.x + w.y * ad.y;
#pragma unroll
    for (int off = 16; off; off >>= 1) {
        p += __shfl_xor(p, off);
        q += __shfl_xor(q, off);
    }
    if (lane == 0) { s_src[row] = p; s_dst[row] = q; }
}

// ---------------------------------------------------------------------------
// Kernel 3: per-edge leaky-relu logits + global max (ordered-uint atomicMax)
// ---------------------------------------------------------------------------
__global__ __launch_bounds__(256) void gat_edge_logit(
    const int* __restrict__ ei, const float* __restrict__ s_src,
    const float* __restrict__ s_dst, float* __restrict__ e_raw,
    unsigned* __restrict__ gmax, int E)
{
    int j = blockIdx.x * blockDim.x + threadIdx.x;
    float v = -3.4e38f;
    if (j < E) {
        int s = ei[j];
        int d = ei[E + j];
        float t = s_src[s] + s_dst[d];
        v = t > 0.0f ? t : NEG_SLOPE * t;
        e_raw[j] = v;
    }
#pragma unroll
    for (int off = 16; off; off >>= 1) {
        float o = __shfl_xor(v, off);
        v = v > o ? v : o;
    }
    if ((threadIdx.x & 31) == 0) atomicMax(gmax, f2ord(v));
}

// ---------------------------------------------------------------------------
// Kernel 4: e = exp(e_raw - gmax) in place, denom[dst] += e
// ---------------------------------------------------------------------------
__global__ __launch_bounds__(256) void gat_exp_denom(
    const int* __restrict__ ei, float* __restrict__ e,
    const unsigned* __restrict__ gmax_u, float* __restrict__ denom, int E)
{
    int j = blockIdx.x * blockDim.x + threadIdx.x;
    if (j >= E) return;
    float g = ord2f(*gmax_u);
    float v = expf(e[j] - g);
    e[j] = v;
    unsafeAtomicAdd(&denom[ei[E + j]], v);
}

// ---------------------------------------------------------------------------
// Kernel 5: out[dst] += Wh[src] * alpha ; one wave per edge, 2 feats per lane
// ---------------------------------------------------------------------------
__global__ __launch_bounds__(256) void gat_scatter(
    const int* __restrict__ ei, const float* __restrict__ Wh,
    const float* __restrict__ e, const float* __restrict__ denom,
    float* __restrict__ out, int E)
{
    long long t = (long long)blockIdx.x * blockDim.x + threadIdx.x;
    int j = (int)(t >> 5);
    if (j >= E) return;
    int lane = (int)(t & 31);
    int s = ei[j];
    int d = ei[E + j];
    float alpha = e[j] / (denom[d] + 1e-16f);
    v2f w = *(const v2f*)(Wh + (size_t)s * OUT_F + 2 * lane);
    float* o = out + (size_t)d * OUT_F + 2 * lane;
    unsafeAtomicAdd(o + 0, w.x * alpha);
    unsafeAtomicAdd(o + 1, w.y * alpha);
}

// ---------------------------------------------------------------------------
// Kernel 6: ELU in place on output
// ---------------------------------------------------------------------------
__global__ __launch_bounds__(256) void gat_elu(float* __restrict__ out, int m)
{
    int i = blockIdx.x * blockDim.x + threadIdx.x;
    if (i >= m) return;
    float v = out[i];
    out[i] = v > 0.0f ? v : (expf(v) - 1.0f);
}

// ---------------------------------------------------------------------------
extern "C" void kernel_launch(void* const* d_in, const int* in_sizes, int n_in,
                              void* d_out, int out_size, void* d_ws, size_t ws_size,
                              hipStream_t stream)
{
    const float* x     = (const float*)d_in[0];
    const int*   ei    = (const int*)d_in[1];     // edge_index [2, E]
    const float* W     = (const float*)d_in[2];
    const float* a_src = (const float*)d_in[3];
    const float* a_dst = (const float*)d_in[4];
    float* out = (float*)d_out;

    int n = in_sizes[0] / IN_F;   // 50000
    int E = in_sizes[1] / 2;      // 1,000,000

    // workspace layout (floats)
    float* ws    = (float*)d_ws;
    float* Wh    = ws;                          // n*64
    float* s_src = Wh + (size_t)n * OUT_F;      // n
    float* s_dst = s_src + n;                   // n
    float* e     = s_dst + n;                   // E
    float* denom = e + E;                       // n
    unsigned* gmax = (unsigned*)(denom + n);    // 1

    // zero-init denom + gmax (ordered-uint 0 == identity for max), and output
    hipMemsetAsync(denom, 0, ((size_t)n + 1) * sizeof(float), stream);
    hipMemsetAsync(out, 0, (size_t)n * OUT_F * sizeof(float), stream);

    // 1. Wh = x @ W  (WMMA fp32)
    gat_gemm_wmma<<<(n + 127) / 128, 256, 0, stream>>>(x, W, Wh, n);

    // 2. node scores
    gat_scores<<<((n * 32) + 255) / 256, 256, 0, stream>>>(Wh, a_src, a_dst, s_src, s_dst, n);

    // 3. edge logits + global max
    gat_edge_logit<<<(E + 255) / 256, 256, 0, stream>>>(ei, s_src, s_dst, e, gmax, E);

    // 4. exp + denominator segment-sum
    gat_exp_denom<<<(E + 255) / 256, 256, 0, stream>>>(ei, e, gmax, denom, E);

    // 5. weighted scatter-add
    long long st = (long long)E * 32;
    gat_scatter<<<(unsigned)((st + 255) / 256), 256, 0, stream>>>(ei, Wh, e, denom, out, E);

    // 6. ELU
    gat_elu<<<((n * OUT_F) + 255) / 256, 256, 0, stream>>>(out, n * OUT_F);
}